// BondAttentionFixed_17798344475006
// MI455X (gfx1250) — compile-verified
//
#include <hip/hip_runtime.h>
#include <cstdint>

typedef __attribute__((ext_vector_type(4))) float v4f;

#define DDIM    256   // feature dim
#define NNODE   2048  // nodes per batch
#define OUTROW  512   // floats per output row (2*DDIM)

// Native f32 atomic add at device scope with an immediate byte offset so all
// atomics in a burst share ONE 64-bit base address register pair.
#define GATOMIC_OFF(p, off, val)                                               \
    asm volatile("global_atomic_add_f32 %0, %1, off offset:" #off              \
                 " scope:SCOPE_DEV"                                            \
                 :: "v"(p), "v"(val) : "memory")

// Adds the 8 floats this lane owns (float4 at +0B, float4 at +512B) into the
// row starting at base, using a single address pair + immediate offsets.
__device__ __forceinline__ void gatomic_row8(float* base, v4f lo, v4f hi) {
    GATOMIC_OFF(base,   0, lo.x);
    GATOMIC_OFF(base,   4, lo.y);
    GATOMIC_OFF(base,   8, lo.z);
    GATOMIC_OFF(base,  12, lo.w);
    GATOMIC_OFF(base, 512, hi.x);
    GATOMIC_OFF(base, 516, hi.y);
    GATOMIC_OFF(base, 520, hi.z);
    GATOMIC_OFF(base, 524, hi.w);
}

// out[row, 0:256) = 0 (regular stores: keep these lines hot in L2 for the atomics)
// out[row, 256:512) = x[row]  (non-temporal: written once, never re-read -> don't
// pollute the 192MB L2 that must hold x + the accumulation half)
__global__ void init_concat_kernel(const v4f* __restrict__ x4,
                                   v4f* __restrict__ out4,
                                   int totalRows) {
    int idx = blockIdx.x * blockDim.x + threadIdx.x;   // one float4 of out
    int row = idx >> 7;                                // 128 float4 per out row
    int c4  = idx & 127;
    if (row >= totalRows) return;
    v4f* dst = out4 + (size_t)row * 128 + c4;
    if (c4 < 64) {
        v4f z = {0.f, 0.f, 0.f, 0.f};
        *dst = z;
    } else {
        v4f v = x4[(size_t)row * 64 + (c4 - 64)];
        __builtin_nontemporal_store(v, dst);
    }
}

// One wave32 per edge. Each lane owns 8 floats of the 256-float row:
// float4 at [4*lane] and float4 at [128 + 4*lane]. All four row loads (both
// directions) are issued before the atomic bursts so VMEM latency overlaps.
__global__ void scatter_edges_kernel(const float* __restrict__ x,
                                     float* __restrict__ out,
                                     const int* __restrict__ bidx,
                                     const int* __restrict__ sidx,
                                     const int* __restrict__ didx,
                                     int E) {
    int gtid = blockIdx.x * blockDim.x + threadIdx.x;
    int edge = gtid >> 5;
    int lane = threadIdx.x & 31;
    if (edge >= E) return;

    // Prefetch the index stream ~64 edges ahead (global_prefetch_b8).
    int pf = edge + 64;
    if (lane == 0 && pf < E) {
        __builtin_prefetch(&bidx[pf], 0, 3);
        __builtin_prefetch(&sidx[pf], 0, 3);
        __builtin_prefetch(&didx[pf], 0, 3);
    }

    int b = bidx[edge];
    int s = sidx[edge];
    int d = didx[edge];

    size_t srow = (size_t)b * NNODE + (size_t)s;
    size_t drow = (size_t)b * NNODE + (size_t)d;

    const v4f* xs = (const v4f*)x + srow * 64;   // 64 float4 = 256 floats
    const v4f* xd = (const v4f*)x + drow * 64;

    // Issue all four 128-bit loads per lane up front.
    v4f a0 = xs[lane];
    v4f a1 = xs[lane + 32];
    v4f c0 = xd[lane];
    v4f c1 = xd[lane + 32];

    float* od = out + drow * OUTROW + 4 * lane;  // dst accumulation half
    float* os = out + srow * OUTROW + 4 * lane;  // src accumulation half

    gatomic_row8(od, a0, a1);   // forward: out[dst] += x[src]
    gatomic_row8(os, c0, c1);   // reverse: out[src] += x[dst]
}

extern "C" void kernel_launch(void* const* d_in, const int* in_sizes, int n_in,
                              void* d_out, int out_size, void* d_ws, size_t ws_size,
                              hipStream_t stream) {
    const float* x    = (const float*)d_in[0];
    const int*   bidx = (const int*)d_in[1];
    const int*   sidx = (const int*)d_in[2];
    const int*   didx = (const int*)d_in[3];
    float*       out  = (float*)d_out;

    int xElems    = in_sizes[0];        // B*N*D = 16,777,216
    int E         = in_sizes[1];        // 1,048,576 edges
    int totalRows = xElems / DDIM;      // B*N = 65,536

    // Pass 1: zero accumulation half + NT-copy concat half (covers entire d_out).
    int outF4   = totalRows * 128;      // float4 elements in out
    int blocks1 = (outF4 + 255) / 256;
    init_concat_kernel<<<blocks1, 256, 0, stream>>>((const v4f*)x, (v4f*)out, totalRows);

    // Pass 2: one wave32 per edge, 8 waves (256 threads) per block.
    int blocks2 = (E + 7) / 8;
    scatter_edges_kernel<<<blocks2, 256, 0, stream>>>(x, out, bidx, sidx, didx, E);
}